// Capsule_60129542149
// MI455X (gfx1250) — compile-verified
//
#include <hip/hip_runtime.h>
#include <math.h>

#define B_BATCH 32
#define L_SEQ   2048
#define D_IN    64
#define NC      32
#define DC      16
#define EPS_F   1e-7f

typedef __attribute__((ext_vector_type(2))) float v2f;
typedef __attribute__((ext_vector_type(8))) float v8f;

// ---------------------------------------------------------------------------
// Phase 1: u_hat[b][n][l][d] = sum_k u[b][l][k] * W[k][n*16+d]
// One wave per (b, ltile(16 l's), ngroup(8 capsules)).  A regs reused x8.
// V_WMMA_F32_16X16X4_F32, chained K=64 -> 16 WMMA per 16x16 tile.
// ---------------------------------------------------------------------------
__global__ void caps_gemm(const float* __restrict__ u,
                          const float* __restrict__ W,
                          float* __restrict__ uhat) {
  const int wave = (blockIdx.x * blockDim.x + threadIdx.x) >> 5;
  const int lane = threadIdx.x & 31;
  const int ng    = wave & 3;          // capsule group: n in [8*ng, 8*ng+8)
  const int ltile = (wave >> 2) & 127; // 128 l-tiles per batch
  const int b     = wave >> 9;

  const int col = lane & 15;           // A: row M; B/D: col N
  const int hi  = lane >> 4;           // low/high half of wave

  // A operand: 16x4 f32 layout. lane<16: {K=4k, 4k+1}; lane>=16: {K=4k+2, 4k+3}
  const float* urow = u + ((size_t)b * L_SEQ + (size_t)ltile * 16 + col) * D_IN;
  const v2f* urow2 = reinterpret_cast<const v2f*>(urow);
  v2f a[16];
#pragma unroll
  for (int k = 0; k < 16; ++k) a[k] = urow2[2 * k + hi];

  for (int n = 8 * ng; n < 8 * ng + 8; ++n) {
    const float* wcol = W + n * DC + col;   // W is [64][512]
    v8f acc = {};
#pragma unroll
    for (int k = 0; k < 16; ++k) {
      const int kk = 4 * k + 2 * hi;
      v2f bm;
      bm.x = wcol[(size_t)kk * (NC * DC)];
      bm.y = wcol[(size_t)(kk + 1) * (NC * DC)];
      acc = __builtin_amdgcn_wmma_f32_16x16x4_f32(
          /*neg_a=*/false, a[k], /*neg_b=*/false, bm,
          /*c_mod=*/(short)0, acc, /*reuse_a=*/false, /*reuse_b=*/false);
    }
    // D layout: VGPR v -> M = v + 8*hi, N = col
    float* out = uhat + (((size_t)b * NC + n) * L_SEQ + (size_t)ltile * 16) * DC;
#pragma unroll
    for (int v = 0; v < 8; ++v) out[(v + 8 * hi) * DC + col] = acc[v];
  }
}

// ---------------------------------------------------------------------------
// Phase 2a: s[b][n][d] = sum_l c[b][n][l] * u_hat[b][n][l][d]; then squash.
// One 256-thread block per (b,n).  float4-coalesced over the 2048x16 slab.
// iter0: c == 1/32 (softmax of zero logits) -> no read of c buffer.
// ---------------------------------------------------------------------------
__global__ void caps_reduce(const float* __restrict__ uhat,
                            const float* __restrict__ c,
                            float* __restrict__ outv, int iter0) {
  const int bn = blockIdx.x;                 // b*NC + n
  const int t  = threadIdx.x;
  const float4* src = reinterpret_cast<const float4*>(uhat + (size_t)bn * L_SEQ * DC);
  const float* cw = c + (size_t)bn * L_SEQ;

  // float4 chunk q covers l = q>>2, d = 4*(q&3)..+3 ; q = t + 256*i keeps q&3 = t&3
  float4 acc = make_float4(0.f, 0.f, 0.f, 0.f);
  const int dg = t & 3;
  for (int q = t; q < (L_SEQ * DC) / 4; q += 256) {
    const int l = q >> 2;
    const float cv = iter0 ? (1.0f / 32.0f) : cw[l];
    float4 v = src[q];
    acc.x += cv * v.x; acc.y += cv * v.y;
    acc.z += cv * v.z; acc.w += cv * v.w;
  }

  __shared__ float part[16][64];
  const int slot = t >> 2;                   // 64 partials per dim
  part[4 * dg + 0][slot] = acc.x;
  part[4 * dg + 1][slot] = acc.y;
  part[4 * dg + 2][slot] = acc.z;
  part[4 * dg + 3][slot] = acc.w;
  __syncthreads();

  __shared__ float sums[16];
  if (t < 16) {
    float s = 0.f;
    for (int i = 0; i < 64; ++i) s += part[t][i];
    sums[t] = s;
  }
  __syncthreads();
  if (t < 16) {
    float ss = EPS_F;
    for (int d = 0; d < 16; ++d) ss += sums[d] * sums[d];
    outv[bn * DC + t] = sums[t] / sqrtf(ss);
  }
}

// ---------------------------------------------------------------------------
// Phase 2b (fused): b_logit[b][n][l] = dot(outputs[b][n][:], u_hat[b][n][l][:])
// then softmax over n -> c[b][n][l].  One thread per l; 256 l's per block.
// ---------------------------------------------------------------------------
__global__ void caps_update(const float* __restrict__ uhat,
                            const float* __restrict__ outv,
                            float* __restrict__ c) {
  const int blk = blockIdx.x;
  const int b   = blk >> 3;                       // 8 chunks of 256 l's per batch
  const int l   = ((blk & 7) << 8) + threadIdx.x;

  __shared__ float ovs[NC * DC];                  // 512 floats, broadcast reads
  ovs[threadIdx.x]       = outv[b * NC * DC + threadIdx.x];
  ovs[threadIdx.x + 256] = outv[b * NC * DC + threadIdx.x + 256];
  __syncthreads();

  float vals[NC];
  float m = -3.402823466e38f;
#pragma unroll 4
  for (int n = 0; n < NC; ++n) {
    const float4* uh4 = reinterpret_cast<const float4*>(
        uhat + (((size_t)b * NC + n) * L_SEQ + l) * DC);
    float dot = 0.f;
#pragma unroll
    for (int q = 0; q < 4; ++q) {
      float4 v = uh4[q];
      dot += v.x * ovs[n * DC + 4 * q + 0];
      dot += v.y * ovs[n * DC + 4 * q + 1];
      dot += v.z * ovs[n * DC + 4 * q + 2];
      dot += v.w * ovs[n * DC + 4 * q + 3];
    }
    vals[n] = dot;
    m = fmaxf(m, dot);
  }
  float sum = 0.f;
#pragma unroll
  for (int n = 0; n < NC; ++n) { vals[n] = expf(vals[n] - m); sum += vals[n]; }
  const float inv = 1.0f / sum;
#pragma unroll
  for (int n = 0; n < NC; ++n)
    c[((size_t)b * NC + n) * L_SEQ + l] = vals[n] * inv;
}

// ---------------------------------------------------------------------------
extern "C" void kernel_launch(void* const* d_in, const int* in_sizes, int n_in,
                              void* d_out, int out_size, void* d_ws, size_t ws_size,
                              hipStream_t stream) {
  const float* u = (const float*)d_in[0];   // [32, 2048, 64]
  const float* W = (const float*)d_in[1];   // [1, 64, 512]
  float* out = (float*)d_out;               // [32, 32, 16]

  float* uhat = (float*)d_ws;                                   // 33,554,432 f32 (128 MB)
  float* cbuf = uhat + (size_t)B_BATCH * NC * L_SEQ * DC;       //  2,097,152 f32 (8 MB)
  float* ov   = cbuf + (size_t)B_BATCH * NC * L_SEQ;            //     16,384 f32

  // GEMM: 16384 waves (32 b x 128 ltiles x 4 ngroups), 8 waves/block
  caps_gemm<<<2048, 256, 0, stream>>>(u, W, uhat);

  // routing iteration 0 (c = 1/32 analytically)
  caps_reduce<<<B_BATCH * NC, 256, 0, stream>>>(uhat, cbuf, ov, 1);
  caps_update<<<B_BATCH * 8, 256, 0, stream>>>(uhat, ov, cbuf);
  // iteration 1
  caps_reduce<<<B_BATCH * NC, 256, 0, stream>>>(uhat, cbuf, ov, 0);
  caps_update<<<B_BATCH * 8, 256, 0, stream>>>(uhat, ov, cbuf);
  // iteration 2 -> final output
  caps_reduce<<<B_BATCH * NC, 256, 0, stream>>>(uhat, cbuf, out, 0);
}